// HGCN_13975823581430
// MI455X (gfx1250) — compile-verified
//
#include <hip/hip_runtime.h>
#include <hip/hip_bf16.h>
#include <math.h>

typedef __attribute__((ext_vector_type(2))) float v2f;
typedef __attribute__((ext_vector_type(8))) float v8f;

#define HID 128

// ------------------------- elementwise / degree kernels -------------------------

__global__ void k_deg_init(float* __restrict__ deg, int n) {
  int i = blockIdx.x * blockDim.x + threadIdx.x;
  if (i < n) deg[i] = 1.0f;  // self-loop contributes 1 to every node's degree
}

__global__ void k_deg_count(const int* __restrict__ dst, float* __restrict__ deg, int E) {
  int i = blockIdx.x * blockDim.x + threadIdx.x;
  int stride = gridDim.x * blockDim.x;
  for (int e = i; e < E; e += stride) atomicAdd(&deg[dst[e]], 1.0f);
}

__global__ void k_rsqrt_inplace(float* __restrict__ v, int n) {
  int i = blockIdx.x * blockDim.x + threadIdx.x;
  if (i < n) v[i] = rsqrtf(v[i]);  // deg >= 1 always (self loops)
}

// agg[i][f] = h[i][f]*dinv[i]^2 + bias[f]   (self-loop term + bias, also zero-inits buffer)
__global__ void k_agg_init(const float* __restrict__ h, const float* __restrict__ dinv,
                           const float* __restrict__ bias, float* __restrict__ agg, int n) {
  int idx = blockIdx.x * blockDim.x + threadIdx.x;
  if (idx < n * HID) {
    int i = idx >> 7;
    int f = idx & (HID - 1);
    float di = dinv[i];
    agg[idx] = h[idx] * di * di + bias[f];
  }
}

__global__ void k_elu(const float* __restrict__ in, float* __restrict__ out, int total) {
  int i = blockIdx.x * blockDim.x + threadIdx.x;
  if (i < total) {
    float x = in[i];
    out[i] = (x > 0.0f) ? x : expm1f(x);
  }
}

// ------------------------- edge scatter (bandwidth-bound core) -------------------------
// 128 threads = one feature each; e is uniform per block -> scalar src/dst/dinv loads.
__global__ void k_edge_scatter(const int* __restrict__ src, const int* __restrict__ dst,
                               const float* __restrict__ dinv, const float* __restrict__ h,
                               float* __restrict__ agg, int E) {
  const int f = threadIdx.x;  // 0..127
  for (int e = blockIdx.x; e < E; e += gridDim.x) {
    const int s = src[e];
    const int d = dst[e];
    const float w = dinv[s] * dinv[d];
    const float val = h[(size_t)s * HID + f] * w;
    atomicAdd(&agg[(size_t)d * HID + f], val);
  }
}

// ------------------------- fp32 WMMA GEMM: C[M x 128] = A[M x K] @ B[K x 128] (+bias) ----
// Block: 256 threads = 8 waves; each wave owns one 16x16 column tile (8*16 = 128 cols).
// A tile (16 x K) staged in LDS with +2 row pad to avoid 64-bank conflicts.
__global__ void k_gemm_wmma(const float* __restrict__ A, const float* __restrict__ B,
                            const float* __restrict__ bias, float* __restrict__ C,
                            int M, int K) {
  extern __shared__ float lds[];
  const int LDSK = K + 2;
  const int row0 = blockIdx.x * 16;
  const int tid = threadIdx.x;

  // cooperative, coalesced load of the 16 x K A tile
  for (int idx = tid; idx < 16 * K; idx += 256) {
    int r = idx / K;
    int c = idx - r * K;
    int row = row0 + r;
    lds[r * LDSK + c] = (row < M) ? A[(size_t)row * K + c] : 0.0f;
  }
  __syncthreads();

  const int lane = tid & 31;
  const int n = ((tid >> 5) << 4) + (lane & 15);  // global output column for this lane
  const int m = lane & 15;                        // A-frag row (ISA 16x4 f32 layout)
  const int kh = (lane >> 4) << 1;                // K sub-offset: lanes 0-15 -> 0, 16-31 -> 2
  const float* __restrict__ Bp = B + n;

  v8f acc = {};
  for (int kk = 0; kk < K; kk += 4) {
    const int kb = kk + kh;
    v2f a;
    a.x = lds[m * LDSK + kb];
    a.y = lds[m * LDSK + kb + 1];
    v2f b;
    b.x = Bp[(size_t)kb * HID];
    b.y = Bp[(size_t)(kb + 1) * HID];
    acc = __builtin_amdgcn_wmma_f32_16x16x4_f32(false, a, false, b, (short)0, acc,
                                                false, false);
  }

  const float bv = bias ? bias[n] : 0.0f;
  const int mbase = (lane >> 4) << 3;  // C/D: vgpr j holds M = j + 8*(lane>=16)
#pragma unroll
  for (int j = 0; j < 8; ++j) {
    int row = row0 + mbase + j;
    if (row < M) C[(size_t)row * HID + n] = acc[j] + bv;
  }
}

// ------------------------- final projection: out = [o1|o2] (M x 256) @ W (256 x 40) + b --
// Block: 96 threads = 3 waves covering columns 0..47; columns >= 40 masked via selects.
__global__ void k_gemm_out(const float* __restrict__ o1, const float* __restrict__ o2,
                           const float* __restrict__ W, const float* __restrict__ bias,
                           float* __restrict__ out, int M, int NOUT) {
  __shared__ float lds[16 * 258];
  const int row0 = blockIdx.x * 16;
  const int tid = threadIdx.x;

  for (int idx = tid; idx < 16 * 256; idx += 96) {
    int r = idx >> 8;
    int c = idx & 255;
    int row = row0 + r;
    float v = 0.0f;
    if (row < M) v = (c < HID) ? o1[(size_t)row * HID + c] : o2[(size_t)row * HID + (c - HID)];
    lds[r * 258 + c] = v;
  }
  __syncthreads();

  const int lane = tid & 31;
  const int n = ((tid >> 5) << 4) + (lane & 15);  // 0..47
  const int m = lane & 15;
  const int kh = (lane >> 4) << 1;
  const bool nok = (n < NOUT);

  v8f acc = {};
  for (int kk = 0; kk < 256; kk += 4) {
    const int kb = kk + kh;
    v2f a;
    a.x = lds[m * 258 + kb];
    a.y = lds[m * 258 + kb + 1];
    v2f b;
    b.x = nok ? W[(size_t)kb * NOUT + n] : 0.0f;
    b.y = nok ? W[(size_t)(kb + 1) * NOUT + n] : 0.0f;
    acc = __builtin_amdgcn_wmma_f32_16x16x4_f32(false, a, false, b, (short)0, acc,
                                                false, false);
  }

  if (nok) {
    const float bv = bias[n];
    const int mbase = (lane >> 4) << 3;
#pragma unroll
    for (int j = 0; j < 8; ++j) {
      int row = row0 + mbase + j;
      if (row < M) out[(size_t)row * NOUT + n] = acc[j] + bv;
    }
  }
}

// ------------------------- host driver -------------------------

extern "C" void kernel_launch(void* const* d_in, const int* in_sizes, int n_in,
                              void* d_out, int out_size, void* d_ws, size_t ws_size,
                              hipStream_t stream) {
  const float* x     = (const float*)d_in[0];
  const int*   edges = (const int*)d_in[1];
  const float* cw1   = (const float*)d_in[2];
  const float* cb1   = (const float*)d_in[3];
  const float* lw1   = (const float*)d_in[4];
  const float* lb1   = (const float*)d_in[5];
  const float* cw2   = (const float*)d_in[6];
  const float* cb2   = (const float*)d_in[7];
  const float* lw2   = (const float*)d_in[8];
  const float* lb2   = (const float*)d_in[9];
  const float* ow    = (const float*)d_in[10];
  const float* ob    = (const float*)d_in[11];

  const int N = in_sizes[0] / 256;     // 100000
  const int E = in_sizes[1] / 2;       // 3200000
  const int NOUT = in_sizes[11];       // 40
  const int* src = edges;
  const int* dst = edges + E;

  // workspace layout (floats): dinv[N] | bufA[N*128] | bufB[N*128] | o1[N*128] | o2[N*128]
  float* ws = (float*)d_ws;
  const size_t nAlign = ((size_t)N + 63) & ~(size_t)63;
  const size_t stride = (size_t)N * HID;
  float* dinv = ws;
  float* bufA = ws + nAlign;
  float* bufB = bufA + stride;
  float* o1   = bufB + stride;
  float* o2   = o1 + stride;

  const int gN    = (N + 255) / 256;
  const int gNF   = (int)((stride + 255) / 256);
  const int gRows = (N + 15) / 16;
  const int smemK256 = 16 * (256 + 2) * (int)sizeof(float);
  const int smemK128 = 16 * (128 + 2) * (int)sizeof(float);
  const int edgeBlocks = 8192;

  // --- degrees / normalization ---
  k_deg_init<<<gN, 256, 0, stream>>>(dinv, N);
  k_deg_count<<<2048, 256, 0, stream>>>(dst, dinv, E);
  k_rsqrt_inplace<<<gN, 256, 0, stream>>>(dinv, N);

  // --- layer 1 conv: h = x @ conv_w1 ; agg = scatter + self-loop + bias ---
  k_gemm_wmma<<<gRows, 256, smemK256, stream>>>(x, cw1, nullptr, bufA, N, 256);
  k_agg_init<<<gNF, 256, 0, stream>>>(bufA, dinv, cb1, bufB, N);
  k_edge_scatter<<<edgeBlocks, HID, 0, stream>>>(src, dst, dinv, bufA, bufB, E);
  k_elu<<<gNF, 256, 0, stream>>>(bufB, bufA, (int)stride);
  k_gemm_wmma<<<gRows, 256, smemK128, stream>>>(bufA, lw1, lb1, o1, N, 128);

  // --- layer 2 conv: h = o1 @ conv_w2 ---
  k_gemm_wmma<<<gRows, 256, smemK128, stream>>>(o1, cw2, nullptr, bufA, N, 128);
  k_agg_init<<<gNF, 256, 0, stream>>>(bufA, dinv, cb2, bufB, N);
  k_edge_scatter<<<edgeBlocks, HID, 0, stream>>>(src, dst, dinv, bufA, bufB, E);
  k_elu<<<gNF, 256, 0, stream>>>(bufB, bufA, (int)stride);
  k_gemm_wmma<<<gRows, 256, smemK128, stream>>>(bufA, lw2, lb2, o2, N, 128);

  // --- output head: out = [o1|o2] @ out_w + out_b ---
  k_gemm_out<<<gRows, 96, 0, stream>>>(o1, o2, ow, ob, (float*)d_out, N, NOUT);
}